// PAAPostProcessor_55929064128528
// MI455X (gfx1250) — compile-verified
//
#include <hip/hip_runtime.h>
#include <hip/hip_bf16.h>
#include <math.h>

// ---------------- constants (match reference) ----------------
static constexpr int   K_PRE      = 1000;   // PRE_NMS_TOP_N
static constexpr int   K_POST     = 100;    // POST_TOP_N
static constexpr float NMS_TH     = 0.6f;
static constexpr float PRE_TH     = 0.05f;
static constexpr float IMGW       = 800.0f;
static constexpr float IMGH       = 800.0f;
static constexpr float SIG_INV    = 40.0f;           // 1/SIGMA, SIGMA=0.025
static constexpr float DWCLAMP    = 4.135166556742356f; // log(1000/16)

// ---------------- workspace layout (per image, in 4-byte words) ----------------
static constexpr int WS_SCORE  = 0;      // float[1024]  sorted top scores (desc)
static constexpr int WS_IDX    = 1024;   // int[1024]    flat idx (loc*C+c)
static constexpr int WS_BOX    = 2048;   // float[1024*4] decoded clipped boxes
static constexpr int WS_LAB    = 6144;   // int[1024]    class label 1..C
static constexpr int WS_VALID  = 7168;   // int[1024]    candidate-valid flag
static constexpr int WS_KEEP   = 8192;   // int[1024]    NMS keep flag
static constexpr int WS_SEL    = 9216;   // int[128]     result -> sorted-list position
static constexpr int WS_STRIDE = 9472;   // bytes: 37888 (16B aligned)

typedef __attribute__((ext_vector_type(2))) float v2f;
typedef __attribute__((ext_vector_type(8))) float v8f;

__device__ __forceinline__ unsigned int f2key(float f) {
    unsigned int u = __float_as_uint(f);
    return (u & 0x80000000u) ? ~u : (u | 0x80000000u);
}
__device__ __forceinline__ float key2f(unsigned int k) {
    unsigned int u = (k & 0x80000000u) ? (k & 0x7FFFFFFFu) : ~k;
    return __uint_as_float(u);
}
__device__ __forceinline__ float sigmoidf(float x) { return 1.0f / (1.0f + expf(-x)); }

// ============================================================================
// K1: per-image top-1000 radix-select + bitonic sort (descending, idx tiebreak)
// ============================================================================
__global__ __launch_bounds__(1024)
void topk_kernel(const float* __restrict__ box_cls, const float* __restrict__ iou_pred,
                 float* __restrict__ ws, int C, int HW) {
    const int n   = blockIdx.x;
    const int tid = threadIdx.x;
    const int M   = C * HW;
    const float* cls  = box_cls + (size_t)n * M;   // layout [C, HW]
    const float* ioup = iou_pred + (size_t)n * HW;

    __shared__ unsigned int hist[2048];
    __shared__ unsigned long long sortbuf[1024];
    __shared__ unsigned int sh_prefix, sh_mask, sh_need;
    __shared__ int sh_cntA, sh_cntT;

    if (tid == 0) { sh_prefix = 0u; sh_mask = 0u; sh_need = K_PRE; }
    __syncthreads();

    const int shifts[3] = {21, 10, 0};
    const int nbits[3]  = {11, 11, 10};

    for (int p = 0; p < 3; ++p) {
        const int shift = shifts[p];
        const unsigned int bmask = (1u << nbits[p]) - 1u;
        for (int b = tid; b < 2048; b += 1024) hist[b] = 0u;
        __syncthreads();
        const unsigned int prefix = sh_prefix, pmask = sh_mask;
        for (int t = tid; t < M; t += 1024) {
            __builtin_prefetch(&cls[t + 1024], 0, 1);   // global_prefetch_b8
            int loc = t % HW;
            float s = sigmoidf(cls[t]);
            float merged = sqrtf(s * sigmoidf(ioup[loc]));
            float m = (s > PRE_TH) ? merged : -1.0f;
            unsigned int key = f2key(m);
            if ((key & pmask) == prefix)
                atomicAdd(&hist[(key >> shift) & bmask], 1u);
        }
        __syncthreads();
        if (tid == 0) {
            unsigned int cum = 0; int chosen = 0;
            for (int b = (int)bmask; b >= 0; --b) {
                unsigned int c = hist[b];
                if (cum + c >= sh_need) { chosen = b; break; }
                cum += c;
            }
            sh_need  -= cum;
            sh_prefix |= ((unsigned int)chosen) << shift;
            sh_mask   |= bmask << shift;
        }
        __syncthreads();
    }

    const unsigned int T     = sh_prefix;          // exact 1000th-largest key
    const unsigned int needT = sh_need;            // how many key==T to take
    const unsigned int baseT = K_PRE - needT;      // == count of key > T

    for (int i = tid; i < 1024; i += 1024) sortbuf[i] = 0ull;
    if (tid == 0) { sh_cntA = 0; sh_cntT = 0; }
    __syncthreads();

    for (int t = tid; t < M; t += 1024) {
        int loc = t % HW, c = t / HW;
        float s = sigmoidf(cls[t]);
        float merged = sqrtf(s * sigmoidf(ioup[loc]));
        float m = (s > PRE_TH) ? merged : -1.0f;
        unsigned int key = f2key(m);
        if (key < T) continue;
        unsigned int flat = (unsigned int)(loc * C + c);
        unsigned long long pair = ((unsigned long long)key << 32) |
                                  (unsigned long long)(~flat);   // lower idx => bigger pair
        if (key > T) {
            int slot = atomicAdd(&sh_cntA, 1);
            sortbuf[slot] = pair;
        } else {
            int tt = atomicAdd(&sh_cntT, 1);
            if ((unsigned int)tt < needT) sortbuf[baseT + tt] = pair;
        }
    }
    __syncthreads();

    // bitonic sort, descending, 1024 elems, one per thread
    for (int k = 2; k <= 1024; k <<= 1) {
        for (int j = k >> 1; j > 0; j >>= 1) {
            int ixj = tid ^ j;
            if (ixj > tid) {
                bool desc = ((tid & k) == 0);
                unsigned long long a = sortbuf[tid], b = sortbuf[ixj];
                if ((a < b) == desc) { sortbuf[tid] = b; sortbuf[ixj] = a; }
            }
            __syncthreads();
        }
    }

    float* wsn = ws + (size_t)n * WS_STRIDE;
    unsigned long long pr = sortbuf[tid];
    unsigned int key  = (unsigned int)(pr >> 32);
    unsigned int flat = ~((unsigned int)(pr & 0xFFFFFFFFull));
    wsn[WS_SCORE + tid] = key2f(key);
    ((int*)wsn)[WS_IDX + tid] = (int)flat;
}

// ============================================================================
// K2: box decode + clip + label/valid
// ============================================================================
__global__ __launch_bounds__(1024)
void decode_kernel(const float* __restrict__ box_regr, const float* __restrict__ anchors,
                   float* __restrict__ ws, int C, int HW) {
    const int n = blockIdx.x, tid = threadIdx.x;
    float* wsn = ws + (size_t)n * WS_STRIDE;
    int*  iwsn = (int*)wsn;
    if (tid >= K_PRE) {   // pad slots 1000..1023
        wsn[WS_BOX + tid*4+0] = 0.f; wsn[WS_BOX + tid*4+1] = 0.f;
        wsn[WS_BOX + tid*4+2] = 0.f; wsn[WS_BOX + tid*4+3] = 0.f;
        iwsn[WS_LAB + tid] = -1; iwsn[WS_VALID + tid] = 0; iwsn[WS_KEEP + tid] = 0;
        wsn[WS_SCORE + tid] = -1.0f;
        return;
    }
    int flat = iwsn[WS_IDX + tid];
    int loc  = flat / C;
    int cl   = flat % C;
    float score = wsn[WS_SCORE + tid];
    const float* rg = box_regr + (size_t)n * 4 * HW;   // [4, HW]
    float p0 = rg[0*HW+loc], p1 = rg[1*HW+loc], p2 = rg[2*HW+loc], p3 = rg[3*HW+loc];
    float a0 = anchors[loc*4+0], a1 = anchors[loc*4+1], a2 = anchors[loc*4+2], a3 = anchors[loc*4+3];
    float w = a2 - a0 + 1.0f, h = a3 - a1 + 1.0f;
    float cx = (a2 + a0) * 0.5f, cy = (a3 + a1) * 0.5f;
    float dx = p0 * 0.1f, dy = p1 * 0.1f;
    float dw = fminf(p2 * 0.2f, DWCLAMP), dh = fminf(p3 * 0.2f, DWCLAMP);
    float pcx = dx * w + cx, pcy = dy * h + cy;
    float pw = expf(dw) * w, ph = expf(dh) * h;
    float x0 = pcx - 0.5f*(pw - 1.0f), y0 = pcy - 0.5f*(ph - 1.0f);
    float x1 = pcx + 0.5f*(pw - 1.0f), y1 = pcy + 0.5f*(ph - 1.0f);
    x0 = fminf(fmaxf(x0, 0.0f), IMGW - 1.0f);
    y0 = fminf(fmaxf(y0, 0.0f), IMGH - 1.0f);
    x1 = fminf(fmaxf(x1, 0.0f), IMGW - 1.0f);
    y1 = fminf(fmaxf(y1, 0.0f), IMGH - 1.0f);
    wsn[WS_BOX + tid*4+0] = x0; wsn[WS_BOX + tid*4+1] = y0;
    wsn[WS_BOX + tid*4+2] = x1; wsn[WS_BOX + tid*4+3] = y1;
    iwsn[WS_LAB + tid]   = cl + 1;
    iwsn[WS_VALID + tid] = (score >= 0.0f) ? 1 : 0;   // == cand mask at selection
}

// ============================================================================
// K3: greedy multi-class NMS (class offset == same-class IoU) + top-100 select
// ============================================================================
__global__ __launch_bounds__(1024)
void nms_kernel(float* __restrict__ ws) {
    const int n = blockIdx.x, tid = threadIdx.x;
    float* wsn = ws + (size_t)n * WS_STRIDE;
    int*  iwsn = (int*)wsn;
    __shared__ float bx0[1024], by0[1024], bx1[1024], by1[1024];
    __shared__ int lab[1024], val[1024], supp[1024], keep[1024];
    __shared__ int curKeep;

    bx0[tid] = wsn[WS_BOX + tid*4+0]; by0[tid] = wsn[WS_BOX + tid*4+1];
    bx1[tid] = wsn[WS_BOX + tid*4+2]; by1[tid] = wsn[WS_BOX + tid*4+3];
    lab[tid] = iwsn[WS_LAB + tid];
    val[tid] = iwsn[WS_VALID + tid];
    supp[tid] = val[tid] ? 0 : 1;   // reference: supp init = !valid
    keep[tid] = 0;
    __syncthreads();

    for (int i = 0; i < K_PRE; ++i) {
        if (tid == 0) {
            int ki = (val[i] && !supp[i]) ? 1 : 0;
            keep[i] = ki; curKeep = ki;
        }
        __syncthreads();
        if (curKeep && tid < K_PRE && lab[tid] == lab[i]) {
            float areaA = (bx1[i]-bx0[i]+1.f)*(by1[i]-by0[i]+1.f);
            float areaB = (bx1[tid]-bx0[tid]+1.f)*(by1[tid]-by0[tid]+1.f);
            float xx0 = fmaxf(bx0[i], bx0[tid]), yy0 = fmaxf(by0[i], by0[tid]);
            float xx1 = fminf(bx1[i], bx1[tid]), yy1 = fminf(by1[i], by1[tid]);
            float iw = fmaxf(xx1-xx0+1.f, 0.f), ih = fmaxf(yy1-yy0+1.f, 0.f);
            float inter = iw*ih;
            float iou = inter / (areaA + areaB - inter);
            if (iou > NMS_TH) supp[tid] = 1;
        }
        __syncthreads();
    }

    iwsn[WS_KEEP + tid] = keep[tid];
    if (tid == 0) {
        // top_k over {kept: score>0, else -1} == kept in order, then non-kept in order
        int cnt = 0;
        for (int j = 0; j < K_PRE && cnt < K_POST; ++j)
            if (keep[j]) iwsn[WS_SEL + cnt++] = j;
        for (int j = 0; j < K_PRE && cnt < K_POST; ++j)
            if (!keep[j]) iwsn[WS_SEL + cnt++] = j;
    }
}

// ============================================================================
// K4: score voting via WMMA f32 16x16x4 (M=112 pad, N=5 pad16, K=1000) + output
//
// Candidate data packed as two float4s (AoS) -> hot loop = 4 x ds_load_b128
// per K-step per lane.  Fused label/valid: cm.z = valid ? label : -2, so the
// mask is a single branch-free float compare.  IoU uses v_rcp_f32 instead of
// the full IEEE division sequence; weight uses hardware v_exp_f32.
// ============================================================================
__device__ __forceinline__ float pis_val(float4 cb, float4 cm,
                                         float rx0, float ry0, float rx1, float ry1,
                                         float rA, float rlabf) {
    float xx0 = fmaxf(rx0, cb.x), yy0 = fmaxf(ry0, cb.y);
    float xx1 = fminf(rx1, cb.z), yy1 = fminf(ry1, cb.w);
    float iw = fmaxf(xx1 - xx0 + 1.f, 0.f), ih = fmaxf(yy1 - yy0 + 1.f, 0.f);
    float inter = iw * ih;
    float den = rA + cm.x - inter;                       // > 0 always
    float iou = inter * __builtin_amdgcn_rcpf(den);      // v_rcp_f32 + v_mul
    float d = 1.0f - iou;
    float w = __expf(-d * d * SIG_INV) * cm.y;           // v_exp_f32
    bool ok = (cm.z == rlabf) & (iou > 0.01f);           // fused label&valid test
    return ok ? w : 0.0f;
}

__device__ __forceinline__ float bsel(float4 cb, int nn) {
    // component nn of augmented row [x0,y0,x1,y1,1,0,...,0]; register-only cndmasks
    float v = cb.x;
    v = (nn == 1) ? cb.y : v;
    v = (nn == 2) ? cb.z : v;
    v = (nn == 3) ? cb.w : v;
    v = (nn == 4) ? 1.0f : v;
    v = (nn >  4) ? 0.0f : v;
    return v;
}

__global__ __launch_bounds__(224)   // 7 wave32 -> 7 M-tiles of 16 rows (112 >= 100)
void vote_kernel(const float* __restrict__ ws, float* __restrict__ out, int N) {
    const int n = blockIdx.x, tid = threadIdx.x;
    const float* wsn = ws + (size_t)n * WS_STRIDE;
    const int*  iwsn = (const int*)wsn;

    __shared__ float4 cbox[1024];    // x0,y0,x1,y1
    __shared__ float4 cmeta[1024];   // area, score, fused label (valid?label:-2), 0
    __shared__ float  rbox[112][4];
    __shared__ int    rlab[112], rkeep[112];
    __shared__ float  rscore[112];
    __shared__ float  vote[112][5];

    const float4* gbox = (const float4*)(wsn + WS_BOX);  // 16B-aligned
    for (int j = tid; j < 1024; j += 224) {
        float4 b = gbox[j];
        cbox[j] = b;
        float4 m;
        m.x = (b.z - b.x + 1.f) * (b.w - b.y + 1.f);     // +1 area convention
        m.y = wsn[WS_SCORE + j];
        m.z = iwsn[WS_VALID + j] ? (float)iwsn[WS_LAB + j] : -2.0f;
        m.w = 0.f;
        cmeta[j] = m;
    }
    for (int m = tid; m < 112; m += 224) {
        if (m < K_POST) {
            int j = iwsn[WS_SEL + m];
            float4 b = gbox[j];
            rbox[m][0] = b.x; rbox[m][1] = b.y; rbox[m][2] = b.z; rbox[m][3] = b.w;
            rlab[m]  = iwsn[WS_LAB + j];
            rkeep[m] = iwsn[WS_KEEP + j];
            rscore[m] = wsn[WS_SCORE + j];
        } else {
            rbox[m][0] = rbox[m][1] = rbox[m][2] = rbox[m][3] = 0.f;
            rlab[m] = -1; rkeep[m] = 0; rscore[m] = -1.f;
        }
    }
    __syncthreads();

    const int wave = tid >> 5;          // 0..6
    const int lane = tid & 31;
    const int half = lane >> 4;         // 0 -> K=k0,k0+1 ; 1 -> K=k0+2,k0+3
    const int nn   = lane & 15;         // A: tile row; B/D: column
    const int mrow = wave * 16 + nn;    // my A row (0..111)

    const float rx0 = rbox[mrow][0], ry0 = rbox[mrow][1];
    const float rx1 = rbox[mrow][2], ry1 = rbox[mrow][3];
    const float rA  = (rx1-rx0+1.f)*(ry1-ry0+1.f);
    const float rlf = (float)rlab[mrow];

    v8f acc = {0.f,0.f,0.f,0.f,0.f,0.f,0.f,0.f};
    for (int k0 = 0; k0 < K_PRE; k0 += 4) {
        const int ka = k0 + half * 2;
        float4 cbA = cbox[ka],   cbB = cbox[ka + 1];     // 2 x ds_load_b128
        float4 cmA = cmeta[ka],  cmB = cmeta[ka + 1];    // 2 x ds_load_b128
        v2f A, B;
        // A (16x4 f32): VGPR0 = K=ka, VGPR1 = K=ka+1 for this half-wave
        A.x = pis_val(cbA, cmA, rx0, ry0, rx1, ry1, rA, rlf);
        A.y = pis_val(cbB, cmB, rx0, ry0, rx1, ry1, rA, rlf);
        // B (4x16 f32): row K=ka (VGPR0) / K=ka+1 (VGPR1), column nn
        B.x = bsel(cbA, nn);
        B.y = bsel(cbB, nn);
        acc = __builtin_amdgcn_wmma_f32_16x16x4_f32(
                  /*neg_a=*/false, A, /*neg_b=*/false, B,
                  /*c_mod=*/(short)0, acc, /*reuse_a=*/false, /*reuse_b=*/false);
    }

    // D layout: VGPR r -> (M = r + 8*half_of_lane, N = nn)
    if (nn < 5) {
        const int mbase = wave * 16 + (half ? 8 : 0);
        #pragma unroll
        for (int r = 0; r < 8; ++r) vote[mbase + r][nn] = acc[r];
    }
    __syncthreads();

    if (tid < K_POST) {
        const int m = tid;
        float denom = vote[m][4];
        int   use   = (rkeep[m] && denom > 0.0f) ? 1 : 0;
        float inv   = 1.0f / fmaxf(denom, 1e-12f);
        float fx0 = use ? vote[m][0]*inv : rbox[m][0];
        float fy0 = use ? vote[m][1]*inv : rbox[m][1];
        float fx1 = use ? vote[m][2]*inv : rbox[m][2];
        float fy1 = use ? vote[m][3]*inv : rbox[m][3];
        size_t base = ((size_t)n * K_POST + m) * 5;
        out[base+0] = fx0; out[base+1] = fy0; out[base+2] = fx1; out[base+3] = fy1;
        out[base+4] = rscore[m];
        out[(size_t)N*K_POST*5 + (size_t)n*K_POST + m]                    = (float)rlab[m];
        out[(size_t)N*K_POST*5 + (size_t)N*K_POST + (size_t)n*K_POST + m] = rkeep[m] ? 1.0f : 0.0f;
    }
}

// ============================================================================
extern "C" void kernel_launch(void* const* d_in, const int* in_sizes, int n_in,
                              void* d_out, int out_size, void* d_ws, size_t ws_size,
                              hipStream_t stream) {
    const float* box_cls  = (const float*)d_in[0];
    const float* box_regr = (const float*)d_in[1];
    const float* iou_pred = (const float*)d_in[2];
    const float* anchors  = (const float*)d_in[3];
    float* out = (float*)d_out;
    float* ws  = (float*)d_ws;

    const int HW = in_sizes[3] / 4;            // anchors: [HW, 4]
    const int N  = in_sizes[2] / HW;           // iou_pred: [N, HW]  (A=1)
    const int C  = in_sizes[0] / (N * HW);     // box_cls: [N, C, HW]

    topk_kernel  <<<N, 1024, 0, stream>>>(box_cls, iou_pred, ws, C, HW);
    decode_kernel<<<N, 1024, 0, stream>>>(box_regr, anchors, ws, C, HW);
    nms_kernel   <<<N, 1024, 0, stream>>>(ws);
    vote_kernel  <<<N,  224, 0, stream>>>(ws, out, N);
}